// FPSampler_42099269435595
// MI455X (gfx1250) — compile-verified
//
#include <hip/hip_runtime.h>
#include <stdint.h>

// ---------------------------------------------------------------------------
// Farthest Point Sampling, MI455X (gfx1250) persistent kernel.
//   B=16 batches, N=131072 points, npoint=1024 sequential rounds.
//   4 WGs per batch (64 WGs), 1024 threads (32 wave32 waves) per WG.
//   - running distance: 32 VGPRs per lane (no spills, never touches memory)
//   - xyz coords: streamed L2 -> LDS each round via Tensor Data Mover
//     (tensor_load_to_lds + s_wait_tensorcnt, double-buffered 128-pt tiles)
//   - argmax: (dist,idx) per lane, packed-u64 __shfl_xor wave32 tree,
//     LDS across waves, 4-WG agent-scope atomic barrier per batch
// NOTE: reference derives the start index from JAX threefry; we use a
// deterministic start=0 per batch (PRNG not reproducible device-side).
// ---------------------------------------------------------------------------

#define B_          16
#define N_          131072
#define NPOINT_MAX  1024
#define WPB         4                    // workgroups per batch
#define NWG         (B_ * WPB)           // 64 workgroups total
#define THREADS     1024
#define WAVES       (THREADS / 32)       // 32 waves
#define CHUNK       (N_ / WPB)           // 32768 points per WG
#define PER_WAVE    (CHUNK / WAVES)      // 1024 points per wave
#define TILE_PTS    128                  // points per TDM tile
#define TILES       (PER_WAVE / TILE_PTS)// 8 tiles per wave per round
#define PL_TILE     (TILE_PTS / 32)      // 4 points per lane per tile
#define DREGS       (PER_WAVE / 32)      // 32 dist registers per lane
#define TILE_DW     (TILE_PTS * 3)       // 384 dwords per tile (fits 16-bit tile_dim0)
#define INF_F       1.0e10f

typedef uint32_t u32;
typedef uint64_t u64;
typedef __attribute__((ext_vector_type(4))) u32 su4;
typedef __attribute__((ext_vector_type(8))) u32 su8;

// Issue one TDM DMA: contiguous TILE_DW dwords, global -> LDS.
// D# per CDNA5 ISA (single-line 2D tile,
// data_size=4B, no pad/iterate/multicast). Group2/3 zero => dims 3/4 unused.
__device__ __forceinline__ void tdm_tile_to_lds(u32 lds_byte_addr, u64 gaddr) {
  su4 g0;
  g0.x = 1u;                                              // count=1 (valid user D#)
  g0.y = lds_byte_addr;                                   // lds_addr
  g0.z = (u32)gaddr;                                      // global_addr lo
  g0.w = ((u32)(gaddr >> 32) & 0x01FFFFFFu) | (2u << 30); // addr[56:32] | type=2
  su8 g1;
  g1.s0 = (2u << 16);                                     // wg_mask=0, data_size=4B
  g1.s1 = ((u32)TILE_DW & 0xFFFFu) << 16;                 // tensor_dim0[15:0]
  g1.s2 = (((u32)TILE_DW >> 16) & 0xFFFFu) | (1u << 16);  // tensor_dim0 hi | tensor_dim1=1
  g1.s3 = ((u32)TILE_DW & 0xFFFFu) << 16;                 // tensor_dim1 hi | tile_dim0
  g1.s4 = 0u;                                             // tile_dim1/2 = 0 (unused)
  g1.s5 = (u32)TILE_DW;                                   // tensor_dim0_stride lo32
  g1.s6 = ((u32)TILE_DW & 0xFFFFu) << 16;                 // stride0 hi | stride1 lo
  g1.s7 = 0u;
  su4 g2 = {0u, 0u, 0u, 0u};
  su4 g3 = {0u, 0u, 0u, 0u};
  asm volatile("tensor_load_to_lds %0, %1, %2, %3"
               :: "s"(g0), "s"(g1), "s"(g2), "s"(g3)
               : "memory");
}

__device__ __forceinline__ u64 u64max(u64 a, u64 b) { return a > b ? a : b; }

__global__ void fps_init_kernel(int* arrive, u64* slots) {
  int t = threadIdx.x;
  if (t < B_) arrive[t] = 0;
  if (t < NWG * 2) slots[t] = 0ull;
}

__global__ __launch_bounds__(THREADS, 1)
void fps_kernel(const float* __restrict__ xyz, const int* __restrict__ npoint_p,
                float* __restrict__ out, int* __restrict__ arrive,
                u64* __restrict__ slots) {
  extern __shared__ char smem[];
  float* tile_lds = (float*)smem;                        // [WAVES][2][TILE_DW]
  u64*   warpKeys = (u64*)(smem + (size_t)WAVES * 2 * TILE_DW * sizeof(float));
  float* bcast    = (float*)(warpKeys + WAVES);          // cx, cy, cz

  const int tid  = threadIdx.x;
  const int lane = tid & 31;
  const int wave = tid >> 5;
  const int wg   = blockIdx.x;
  const int b    = wg / WPB;
  const int wgin = wg % WPB;
  const int npoint = min(npoint_p[0], NPOINT_MAX);

  const float* bxyz = xyz + (size_t)b * N_ * 3;
  const int sliceBase = wgin * CHUNK + wave * PER_WAVE;  // first point (in batch) of this wave

  float* buf0 = tile_lds + (size_t)wave * 2 * TILE_DW;
  float* buf1 = buf0 + TILE_DW;
  const u32 lds0 = (u32)(uintptr_t)buf0;                 // low 32b of flat == LDS offset
  const u32 lds1 = (u32)(uintptr_t)buf1;

  float dist[DREGS];
#pragma unroll
  for (int j = 0; j < DREGS; ++j) dist[j] = INF_F;

  // Round 0 centroid = start point (deterministic start = 0).
  if (tid == 0) {
    bcast[0] = bxyz[0]; bcast[1] = bxyz[1]; bcast[2] = bxyz[2];
    if (wgin == 0) {
      size_t o3 = (size_t)b * npoint * 3;
      out[o3 + 0] = bxyz[0]; out[o3 + 1] = bxyz[1]; out[o3 + 2] = bxyz[2];
      out[(size_t)B_ * npoint * 3 + (size_t)b * npoint] = 0.0f; // idx[0] = start
    }
  }
  __syncthreads();

  for (int it = 0; it < npoint; ++it) {
    const float ccx = bcast[0], ccy = bcast[1], ccz = bcast[2];

    // Prime the TDM pipeline: tile 0 -> buf0.
    tdm_tile_to_lds(lds0, (u64)(uintptr_t)(bxyz + (size_t)sliceBase * 3));

    float bval = -1.0f;
    u32   bidx = 0u;
#pragma unroll
    for (int t = 0; t < TILES; ++t) {
      if (t + 1 < TILES) {  // prefetch next tile, then wait for current one
        const u64 ga = (u64)(uintptr_t)(bxyz + (size_t)(sliceBase + (t + 1) * TILE_PTS) * 3);
        tdm_tile_to_lds(((t + 1) & 1) ? lds1 : lds0, ga);
        __builtin_amdgcn_s_wait_tensorcnt(1);
      } else {
        __builtin_amdgcn_s_wait_tensorcnt(0);
      }
      const float* fb = (t & 1) ? buf1 : buf0;
#pragma unroll
      for (int j = 0; j < PL_TILE; ++j) {
        const int pi = j * 32 + lane;                    // lane-strided: bank conflict-free
        const float x = fb[3 * pi + 0];
        const float y = fb[3 * pi + 1];
        const float z = fb[3 * pi + 2];
        const float dx = x - ccx, dy = y - ccy, dz = z - ccz;
        const float d = dx * dx + dy * dy + dz * dz;
        const int r = t * PL_TILE + j;
        const float nd = fminf(dist[r], d);
        dist[r] = nd;
        if (nd > bval) {                                 // ascending order => first max kept
          bval = nd;
          bidx = (u32)(sliceBase + t * TILE_PTS + pi);
        }
      }
    }

    // Pack once per lane: (dist_bits<<32) | ~idx  (monotone for dist>=0,
    // smallest-index tie break, matching jnp.argmax first-occurrence).
    u64 bestkey = ((u64)__float_as_uint(bval) << 32) | (u64)(0xFFFFFFFFu - bidx);
#pragma unroll
    for (int off = 16; off > 0; off >>= 1)
      bestkey = u64max(bestkey, (u64)__shfl_xor((unsigned long long)bestkey, off, 32));
    if (lane == 0) warpKeys[wave] = bestkey;
    __syncthreads();   // (A) warpKeys ready; everyone done reading bcast

    if (wave == 0) {
      u64 k = warpKeys[lane];  // WAVES == 32 == warpSize
#pragma unroll
      for (int off = 16; off > 0; off >>= 1)
        k = u64max(k, (u64)__shfl_xor((unsigned long long)k, off, 32));
      if (lane == 0) {
        const int phase = it & 1;
        __hip_atomic_store(&slots[wg * 2 + phase], k,
                           __ATOMIC_RELEASE, __HIP_MEMORY_SCOPE_AGENT);
        __hip_atomic_fetch_add(&arrive[b], 1,
                               __ATOMIC_ACQ_REL, __HIP_MEMORY_SCOPE_AGENT);
        const int target = WPB * (it + 1);               // monotonic arrive counter
        while (__hip_atomic_load(&arrive[b], __ATOMIC_ACQUIRE,
                                 __HIP_MEMORY_SCOPE_AGENT) < target)
          __builtin_amdgcn_s_sleep(1);
        u64 kw = 0ull;
#pragma unroll
        for (int w = 0; w < WPB; ++w)
          kw = u64max(kw, __hip_atomic_load(&slots[(b * WPB + w) * 2 + phase],
                                            __ATOMIC_ACQUIRE, __HIP_MEMORY_SCOPE_AGENT));
        const u32 widx = 0xFFFFFFFFu - (u32)kw;          // next farthest (L2-hot gather)
        const float wx = bxyz[3 * widx + 0];
        const float wy = bxyz[3 * widx + 1];
        const float wz = bxyz[3 * widx + 2];
        bcast[0] = wx; bcast[1] = wy; bcast[2] = wz;
        if (wgin == 0 && it + 1 < npoint) {
          const size_t o3 = ((size_t)b * npoint + (it + 1)) * 3;
          out[o3 + 0] = wx; out[o3 + 1] = wy; out[o3 + 2] = wz;
          out[(size_t)B_ * npoint * 3 + (size_t)b * npoint + (it + 1)] = (float)widx;
        }
      }
    }
    __syncthreads();   // (B) bcast holds next centroid for all waves
  }
}

extern "C" void kernel_launch(void* const* d_in, const int* in_sizes, int n_in,
                              void* d_out, int out_size, void* d_ws, size_t ws_size,
                              hipStream_t stream) {
  (void)in_sizes; (void)n_in; (void)out_size; (void)ws_size;
  const float* xyz    = (const float*)d_in[0];
  const int*   npoint = (const int*)d_in[1];
  float*       out    = (float*)d_out;

  int* arrive = (int*)d_ws;                    // 16 ints
  u64* slots  = (u64*)((char*)d_ws + 128);     // 128 u64 (double-buffered)

  fps_init_kernel<<<1, 128, 0, stream>>>(arrive, slots);

  const size_t smem = (size_t)WAVES * 2 * TILE_DW * sizeof(float)   // 96 KB tiles
                    + (size_t)WAVES * sizeof(u64)                   // warp keys
                    + 4 * sizeof(float);                            // centroid bcast
  fps_kernel<<<NWG, THREADS, smem, stream>>>(xyz, npoint, out, arrive, slots);
}